// GCN_IDS_50637664420305
// MI455X (gfx1250) — compile-verified
//
#include <hip/hip_runtime.h>
#include <math.h>

// ---------------------------------------------------------------------------
// GCN (3x GCNConv + BN + ReLU, then MLP head + log_softmax) for MI455X/gfx1250
//
// Compile-only reasoning:
//  * GEMMs total ~3.7 GFLOP fp32 -> nowhere near the matrix ceiling; the cost
//    is data movement. Edge gather/scatter (3 x ~820 MB) dominates, but the
//    25.6 MB activation buffers live in the 192 MB L2, so this is an
//    L2/atomic-bound problem. Hence: fp32 WMMA (exact) instead of bf16.
//  * b0/b1/b2 cancel exactly under the following BatchNorm (column shift),
//    so they are skipped.
// ---------------------------------------------------------------------------

#define GCN_N   100000
#define GCN_E   1600000
#define GCN_IN  128
#define GCN_HID 64
#define BN_BLOCKS 200

typedef float v2f __attribute__((ext_vector_type(2)));
typedef float v8f __attribute__((ext_vector_type(8)));

// ---------------------------------------------------------------------------
// fp32 WMMA GEMM:  C[M,NC] = A[M,K] * W[K,NC]   (+bias, ReLU when EPILOGUE)
// One wave computes a 16-row strip across all NC columns (NC/16 accumulators),
// reusing the A fragment for every N-tile. M must be a multiple of 16
// (100000 = 6250*16). Tile guard is wave-uniform so EXEC is all-ones at WMMA.
//
// V_WMMA_F32_16X16X4_F32 VGPR layouts (ISA 7.12.2, wave32):
//   A (16x4):  lanes 0-15 -> K=0,1 ; lanes 16-31 -> K=2,3 (VGPR = K lsb)
//   B (4x16):  lane half selects K-half, VGPR index = K lsb within half
//   C/D:       VGPR i = row i (lanes 0-15) / row i+8 (lanes 16-31)
// ---------------------------------------------------------------------------
template <int K, int NC, bool EPILOGUE>
__global__ void gemm_wmma_f32(const float* __restrict__ A,
                              const float* __restrict__ W,
                              float* __restrict__ C,
                              const float* __restrict__ bias,
                              int M) {
  const int wave = blockIdx.x * (blockDim.x >> 5) + (threadIdx.x >> 5);
  const int lane = threadIdx.x & 31;
  if (wave * 16 >= M) return;                 // uniform per wave -> EXEC all 1s

  const int mlane = lane & 15;                // M (for A) / N (for B,C) lane idx
  const int khalf = (lane >> 4) * 2;          // 0 or 2: K sub-block per lane half
  const int row   = wave * 16 + mlane;

  v8f acc[NC / 16];
#pragma unroll
  for (int nt = 0; nt < NC / 16; ++nt) acc[nt] = (v8f)0.0f;

  for (int k0 = 0; k0 < K; k0 += 4) {
    // A fragment: 2 consecutive fp32 at even K offset -> aligned 8B load
    const v2f a = *(const v2f*)(A + (size_t)row * K + k0 + khalf);
#pragma unroll
    for (int nt = 0; nt < NC / 16; ++nt) {
      const int col = nt * 16 + mlane;
      v2f b;
      b.x = W[(size_t)(k0 + khalf) * NC + col];
      b.y = W[(size_t)(k0 + khalf + 1) * NC + col];
      acc[nt] = __builtin_amdgcn_wmma_f32_16x16x4_f32(
          /*neg_a=*/false, a, /*neg_b=*/false, b,
          /*c_mod=*/(short)0, acc[nt], /*reuse_a=*/false, /*reuse_b=*/false);
    }
  }

  const int rbase = wave * 16 + ((lane >> 4) ? 8 : 0);
#pragma unroll
  for (int nt = 0; nt < NC / 16; ++nt) {
    const int col = nt * 16 + mlane;
#pragma unroll
    for (int i = 0; i < 8; ++i) {
      float v = acc[nt][i];
      if (EPILOGUE) {
        v += bias[col];
        v = v > 0.0f ? v : 0.0f;
      }
      C[(size_t)(rbase + i) * NC + col] = v;
    }
  }
}

// ---------------------------------------------------------------------------
// Degree / normalization
// ---------------------------------------------------------------------------
__global__ void zero_i32(int* __restrict__ p, int n) {
  int i = blockIdx.x * blockDim.x + threadIdx.x;
  if (i < n) p[i] = 0;
}

__global__ void deg_count(const int* __restrict__ dst, int* __restrict__ degi, int e) {
  int i = blockIdx.x * blockDim.x + threadIdx.x;
  if (i < e) atomicAdd(&degi[dst[i]], 1);
}

__global__ void make_dis(const int* __restrict__ degi, float* __restrict__ dis, int n) {
  int i = blockIdx.x * blockDim.x + threadIdx.x;
  if (i < n) dis[i] = rsqrtf((float)degi[i] + 1.0f);   // deg includes self-loop
}

// ---------------------------------------------------------------------------
// agg init with self-loop term: agg = h * dis^2  (elementwise over N*HID)
// ---------------------------------------------------------------------------
__global__ void selfloop_init(const float* __restrict__ h,
                              const float* __restrict__ dis,
                              float* __restrict__ agg, int n) {
  int i = blockIdx.x * blockDim.x + threadIdx.x;
  if (i < n) {
    float d = dis[i >> 6];     // HID = 64
    agg[i] = h[i] * d * d;
  }
}

// ---------------------------------------------------------------------------
// Edge scatter: agg[dst] += h[src] * dis[src]*dis[dst]
// 16 threads per edge, float4 per thread; h/agg are L2-resident (25.6 MB).
// ---------------------------------------------------------------------------
__global__ void gcn_scatter(const float* __restrict__ h,
                            const int* __restrict__ src,
                            const int* __restrict__ dst,
                            const float* __restrict__ dis,
                            float* __restrict__ agg, int e) {
  int t = blockIdx.x * blockDim.x + threadIdx.x;   // e*16 = 25.6M < 2^31
  if (t >= e * 16) return;
  const int  ed = t >> 4;
  const int  c  = (t & 15) * 4;
  const int  s  = src[ed];
  const int  d  = dst[ed];
  const float norm = dis[s] * dis[d];
  const float4 hv = *(const float4*)(h + (size_t)s * 64 + c);
  float* ap = agg + (size_t)d * 64 + c;
  atomicAdd(ap + 0, hv.x * norm);
  atomicAdd(ap + 1, hv.y * norm);
  atomicAdd(ap + 2, hv.z * norm);
  atomicAdd(ap + 3, hv.w * norm);
}

// ---------------------------------------------------------------------------
// BatchNorm: deterministic two-stage column reduction (64 cols), then apply.
// ---------------------------------------------------------------------------
__global__ void bn_partial(const float* __restrict__ x,
                           float* __restrict__ partial, int n) {
  __shared__ float s_sum[256];
  __shared__ float s_sq[256];
  const int col = threadIdx.x & 63;
  const int rg  = threadIdx.x >> 6;                 // 0..3
  const int rows_per_block = (n + gridDim.x - 1) / gridDim.x;
  const int r0 = blockIdx.x * rows_per_block;
  const int r1 = min(n, r0 + rows_per_block);
  float s = 0.0f, q = 0.0f;
  for (int r = r0 + rg; r < r1; r += 4) {
    float v = x[(size_t)r * 64 + col];
    s += v;
    q += v * v;
  }
  s_sum[threadIdx.x] = s;
  s_sq[threadIdx.x]  = q;
  __syncthreads();
  if (rg == 0) {
    s = s_sum[col] + s_sum[col + 64] + s_sum[col + 128] + s_sum[col + 192];
    q = s_sq[col]  + s_sq[col + 64]  + s_sq[col + 128]  + s_sq[col + 192];
    partial[(size_t)blockIdx.x * 128 + col]      = s;
    partial[(size_t)blockIdx.x * 128 + 64 + col] = q;
  }
}

__global__ void bn_finalize(const float* __restrict__ partial, int nparts,
                            float* __restrict__ stats, int n) {
  const int col = threadIdx.x;                      // 64 threads
  float s = 0.0f, q = 0.0f;
  for (int b = 0; b < nparts; ++b) {
    s += partial[(size_t)b * 128 + col];
    q += partial[(size_t)b * 128 + 64 + col];
  }
  const float mu  = s / (float)n;
  const float var = q / (float)n - mu * mu;         // biased variance
  stats[col]      = mu;
  stats[64 + col] = rsqrtf(var + 1e-5f);
}

__global__ void bn_apply_relu(float* __restrict__ x,
                              const float* __restrict__ stats,
                              const float* __restrict__ gamma,
                              const float* __restrict__ beta, int n) {
  int i = blockIdx.x * blockDim.x + threadIdx.x;
  if (i >= n) return;
  const int col = i & 63;
  float v = (x[i] - stats[col]) * stats[64 + col] * gamma[col] + beta[col];
  x[i] = v > 0.0f ? v : 0.0f;
}

// ---------------------------------------------------------------------------
// Head: out = log_softmax(z1 @ fc2_w + fc2_b), z1 is [N,32], fc2_w is [32,2]
// ---------------------------------------------------------------------------
__global__ void head_kernel(const float* __restrict__ z1,
                            const float* __restrict__ fc2_w,
                            const float* __restrict__ fc2_b,
                            float* __restrict__ out, int n) {
  int r = blockIdx.x * blockDim.x + threadIdx.x;
  if (r >= n) return;
  float a = fc2_b[0], b = fc2_b[1];
  const float* zr = z1 + (size_t)r * 32;
#pragma unroll
  for (int k = 0; k < 32; ++k) {
    const float v = zr[k];
    a += v * fc2_w[2 * k];
    b += v * fc2_w[2 * k + 1];
  }
  const float m   = fmaxf(a, b);
  const float lse = m + logf(expf(a - m) + expf(b - m));
  out[(size_t)r * 2]     = a - lse;
  out[(size_t)r * 2 + 1] = b - lse;
}

// ---------------------------------------------------------------------------
// Launch
// ---------------------------------------------------------------------------
extern "C" void kernel_launch(void* const* d_in, const int* in_sizes, int n_in,
                              void* d_out, int out_size, void* d_ws, size_t ws_size,
                              hipStream_t stream) {
  const int N = GCN_N, E = GCN_E;

  const float* x     = (const float*)d_in[0];
  const int*   edge  = (const int*)d_in[1];
  const float* W0    = (const float*)d_in[2];
  // d_in[3] (b0), d_in[5] (b1), d_in[7] (b2): cancel exactly under BatchNorm.
  const float* W1    = (const float*)d_in[4];
  const float* W2    = (const float*)d_in[6];
  const float* g0    = (const float*)d_in[8];
  const float* be0   = (const float*)d_in[9];
  const float* g1    = (const float*)d_in[10];
  const float* be1   = (const float*)d_in[11];
  const float* g2    = (const float*)d_in[12];
  const float* be2   = (const float*)d_in[13];
  const float* fc1_w = (const float*)d_in[14];
  const float* fc1_b = (const float*)d_in[15];
  const float* fc2_w = (const float*)d_in[16];
  const float* fc2_b = (const float*)d_in[17];
  float* out = (float*)d_out;

  const int* src = edge;        // edge_index row 0
  const int* dst = edge + E;    // edge_index row 1

  // Workspace carve (all chunks 16B aligned; total ~77.8 MB)
  char* w = (char*)d_ws;
  int*   degi    = (int*)w;           w += (size_t)N * sizeof(int);
  float* dis     = (float*)w;         w += (size_t)N * sizeof(float);
  float* bufH    = (float*)w;         w += (size_t)N * 64 * sizeof(float);
  float* bufA    = (float*)w;         w += (size_t)N * 64 * sizeof(float);
  float* bufB    = (float*)w;         w += (size_t)N * 64 * sizeof(float);
  float* partial = (float*)w;         w += (size_t)BN_BLOCKS * 128 * sizeof(float);
  float* stats   = (float*)w;         w += 128 * sizeof(float);

  const int gN     = (N + 255) / 256;            // 391
  const int gNH    = (N * 64 + 255) / 256;       // 25000
  const int gE     = (E + 255) / 256;            // 6250
  const int gScat  = (E * 16) / 256;             // 100000
  const int gGemm  = ((N / 16) + 7) / 8;         // 782 blocks, 8 waves each

  // Degree -> dis = rsqrt(deg + 1)
  zero_i32<<<gN, 256, 0, stream>>>(degi, N);
  deg_count<<<gE, 256, 0, stream>>>(dst, degi, E);
  make_dis<<<gN, 256, 0, stream>>>(degi, dis, N);

  // ---- Layer 0: x[N,128] -> bufA[N,64] ----
  gemm_wmma_f32<GCN_IN, GCN_HID, false><<<gGemm, 256, 0, stream>>>(x, W0, bufH, nullptr, N);
  selfloop_init<<<gNH, 256, 0, stream>>>(bufH, dis, bufA, N * 64);
  gcn_scatter<<<gScat, 256, 0, stream>>>(bufH, src, dst, dis, bufA, E);
  bn_partial<<<BN_BLOCKS, 256, 0, stream>>>(bufA, partial, N);
  bn_finalize<<<1, 64, 0, stream>>>(partial, BN_BLOCKS, stats, N);
  bn_apply_relu<<<gNH, 256, 0, stream>>>(bufA, stats, g0, be0, N * 64);

  // ---- Layer 1: bufA -> bufB ----
  gemm_wmma_f32<GCN_HID, GCN_HID, false><<<gGemm, 256, 0, stream>>>(bufA, W1, bufH, nullptr, N);
  selfloop_init<<<gNH, 256, 0, stream>>>(bufH, dis, bufB, N * 64);
  gcn_scatter<<<gScat, 256, 0, stream>>>(bufH, src, dst, dis, bufB, E);
  bn_partial<<<BN_BLOCKS, 256, 0, stream>>>(bufB, partial, N);
  bn_finalize<<<1, 64, 0, stream>>>(partial, BN_BLOCKS, stats, N);
  bn_apply_relu<<<gNH, 256, 0, stream>>>(bufB, stats, g1, be1, N * 64);

  // ---- Layer 2: bufB -> bufA ----
  gemm_wmma_f32<GCN_HID, GCN_HID, false><<<gGemm, 256, 0, stream>>>(bufB, W2, bufH, nullptr, N);
  selfloop_init<<<gNH, 256, 0, stream>>>(bufH, dis, bufA, N * 64);
  gcn_scatter<<<gScat, 256, 0, stream>>>(bufH, src, dst, dis, bufA, E);
  bn_partial<<<BN_BLOCKS, 256, 0, stream>>>(bufA, partial, N);
  bn_finalize<<<1, 64, 0, stream>>>(partial, BN_BLOCKS, stats, N);
  bn_apply_relu<<<gNH, 256, 0, stream>>>(bufA, stats, g2, be2, N * 64);

  // ---- FC1 (bias + ReLU fused into WMMA epilogue): bufA[N,64] -> bufH[N,32]
  gemm_wmma_f32<GCN_HID, 32, true><<<gGemm, 256, 0, stream>>>(bufA, fc1_w, bufH, fc1_b, N);

  // ---- FC2 + log_softmax ----
  head_kernel<<<gN, 256, 0, stream>>>(bufH, fc2_w, fc2_b, out, N);
}